// Spacial_IRNN_15264313770032
// MI455X (gfx1250) — compile-verified
//
#include <hip/hip_runtime.h>
#include <stdint.h>

// Problem dims (fixed by the reference)
#define B_ 8
#define C_ 32
#define H_ 256
#define W_ 256
#define NPLANE (H_ * W_)
#define NTOT   (B_ * C_ * H_ * W_)

// Horizontal-scan tiling
#define ROWS 128          // rows per workgroup tile (one thread per row)
#define LDS_PITCH 258     // 256 + 2 pad dwords (inserted by TDM):
                          //  - scan phase: bank = (2t+w)%64, conflict-free
                          //  - rows 8B-aligned -> float2 writeback path

typedef unsigned int v4u __attribute__((ext_vector_type(4)));
typedef int          v8i __attribute__((ext_vector_type(8)));
typedef int          v4i __attribute__((ext_vector_type(4)));

// ---------------------------------------------------------------------------
// Vertical scans (top-down + bottom-up). Threads map to (b,c,w): W is the
// contiguous axis, so every global load/store is a fully coalesced 128B line
// per wave32. Both directions in one kernel; the second pass re-reads x from
// the 192MB L2.
// ---------------------------------------------------------------------------
__global__ void irnn_vertical(const float* __restrict__ x,
                              const float* __restrict__ alpha,
                              float* __restrict__ out_td,
                              float* __restrict__ out_dt) {
  int gid = blockIdx.x * blockDim.x + threadIdx.x;   // 0 .. B*C*W-1
  int bc  = gid >> 8;                                // plane index b*C + c
  float a = alpha[bc & (C_ - 1)];
  long base = (long)bc * NPLANE + (gid & (W_ - 1));

  float prev = 0.f;
#pragma unroll 4
  for (int h = 0; h < H_; ++h) {
    long idx = base + ((long)h << 8);
    __builtin_prefetch(&x[idx + (8 << 8)], 0, 3);    // global_prefetch_b8, WGP scope
    float cur = fmaxf(fmaf(a, prev, x[idx]), 0.f);
    out_td[idx] = cur;
    prev = cur;
  }
  prev = 0.f;
#pragma unroll 4
  for (int h = H_ - 1; h >= 0; --h) {
    long idx = base + ((long)h << 8);
    float cur = fmaxf(fmaf(a, prev, x[idx]), 0.f);
    out_dt[idx] = cur;
    prev = cur;
  }
}

// ---------------------------------------------------------------------------
// TDM: DMA a ROWS x 256 fp32 tile (row pitch 256 elements in memory) into LDS,
// inserting 2 pad dwords after every 256 dwords -> LDS row pitch = 258 floats.
// ---------------------------------------------------------------------------
__device__ static inline void tdm_load_tile(const float* gsrc, uint32_t lds_off) {
  uint64_t ga = (uint64_t)(uintptr_t)gsrc;
  v4u g0 = {
    1u,                                                  // count=1 (valid D#)
    lds_off,                                             // lds_addr (bytes)
    (uint32_t)ga,                                        // global_addr[31:0]
    ((uint32_t)(ga >> 32) & 0x1FFFFFFu) | 0x80000000u    // addr[56:32] | type=2
  };
  v8i g1 = {
    (int)((2u << 16) | (1u << 20) | (7u << 22) | (1u << 25)),
                                                  // data_size=4B, pad_enable,
                                                  // pad_interval=256dw, pad=2dw
    (int)(256u << 16),                            // tensor_dim0 = 256
    0,                                            // tensor_dim1[15:0] (65536->0)
    (int)((256u << 16) | 1u),                     // tile_dim0=256, dim1[31:16]=1
    ROWS,                                         // tile_dim1 = 128 rows
    256,                                          // tensor_dim0_stride = 256
    0, 0                                          // dim1_stride = 0 (2-D)
  };
  v4i z4 = {0, 0, 0, 0};
#if defined(__clang_major__) && (__clang_major__ >= 23)
  v8i z8 = {0, 0, 0, 0, 0, 0, 0, 0};
  __builtin_amdgcn_tensor_load_to_lds(g0, g1, z4, z4, z8, 0);
#else
  __builtin_amdgcn_tensor_load_to_lds(g0, g1, z4, z4, 0);
#endif
}

// ---------------------------------------------------------------------------
// Horizontal scans (left->right + right->left). 128 threads, each owns one
// row of the LDS tile. TDM load -> in-place scan -> float2 coalesced store;
// TDM reload of the pristine tile (L2 hit) -> reverse scan -> store.
// ---------------------------------------------------------------------------
__global__ void irnn_horizontal(const float* __restrict__ x,
                                const float* __restrict__ alpha,
                                float* __restrict__ out_lr,
                                float* __restrict__ out_rl) {
  extern __shared__ float tile[];   // ROWS * LDS_PITCH floats = 132096 B
  const int  tid     = threadIdx.x;                   // 0..127
  const long rowBase = (long)blockIdx.x * ROWS;       // flattened (b,c,h) row
  const float a      = alpha[(int)((rowBase >> 8) & (C_ - 1))];
  const float* gsrc  = x + rowBase * W_;
  const uint32_t lds_off = (uint32_t)(uintptr_t)tile;

  // ---- pass 1: left -> right -------------------------------------------
  if ((tid >> 5) == 0) {                 // wave 0 issues the DMA
    tdm_load_tile(gsrc, lds_off);
    __builtin_amdgcn_s_wait_tensorcnt(0);
  }
  __syncthreads();

  {
    float* row = tile + tid * LDS_PITCH;
    float prev = 0.f;
    for (int w = 0; w < W_; ++w) {
      float cur = fmaxf(fmaf(a, prev, row[w]), 0.f);
      row[w] = cur;                      // in-place: thread owns this row
      prev = cur;
    }
  }
  __syncthreads();

  {
    // float2 writeback: 8B-aligned LDS rows (pitch 258), b64 global stores
    float2* dst2 = (float2*)(out_lr + rowBase * W_);
    for (int i = tid; i < ROWS * (W_ / 2); i += blockDim.x) {
      int r = i >> 7, j = i & ((W_ / 2) - 1);
      dst2[i] = ((const float2*)(tile + r * LDS_PITCH))[j];
    }
  }
  __syncthreads();

  // ---- pass 2: right -> left (refetch pristine x tile; hits L2) --------
  if ((tid >> 5) == 0) {
    tdm_load_tile(gsrc, lds_off);
    __builtin_amdgcn_s_wait_tensorcnt(0);
  }
  __syncthreads();

  {
    float* row = tile + tid * LDS_PITCH;
    float prev = 0.f;
    for (int w = W_ - 1; w >= 0; --w) {
      float cur = fmaxf(fmaf(a, prev, row[w]), 0.f);
      row[w] = cur;
      prev = cur;
    }
  }
  __syncthreads();

  {
    float2* dst2 = (float2*)(out_rl + rowBase * W_);
    for (int i = tid; i < ROWS * (W_ / 2); i += blockDim.x) {
      int r = i >> 7, j = i & ((W_ / 2) - 1);
      dst2[i] = ((const float2*)(tile + r * LDS_PITCH))[j];
    }
  }
}

// ---------------------------------------------------------------------------
extern "C" void kernel_launch(void* const* d_in, const int* in_sizes, int n_in,
                              void* d_out, int out_size, void* d_ws, size_t ws_size,
                              hipStream_t stream) {
  const float* x     = (const float*)d_in[0];
  const float* alpha = (const float*)d_in[1];
  float* out = (float*)d_out;

  const long N = (long)NTOT;
  float* h_td = out;           // output order: (h_td, h_lr, h_dt, h_rl)
  float* h_lr = out + N;
  float* h_dt = out + 2 * N;
  float* h_rl = out + 3 * N;

  // Vertical scans: 65536 threads, one per (b,c,w) column.
  irnn_vertical<<<dim3((B_ * C_ * W_) / 256), dim3(256), 0, stream>>>(
      x, alpha, h_td, h_dt);

  // Horizontal scans: 512 workgroups x 128 threads, TDM-staged LDS tiles.
  size_t shmem = (size_t)ROWS * LDS_PITCH * sizeof(float);
  irnn_horizontal<<<dim3((B_ * C_ * H_) / ROWS), dim3(ROWS), shmem, stream>>>(
      x, alpha, h_lr, h_rl);
}